// EdgeGAT3Block_73220602462596
// MI455X (gfx1250) — compile-verified
//
#include <hip/hip_runtime.h>
#include <hip/hip_bf16.h>

typedef __attribute__((ext_vector_type(2))) float v2f;
typedef __attribute__((ext_vector_type(8))) float v8f;

#define N_NODES 50000
#define N_EDGES 800000
#define HID     64
#define HEADS   4
#define HC      256   // HID*HEADS
#define EDIM    32
#define NEG_SLOPE 0.2f
#define LN_EPS  1e-5f

// ---- monotone float<->uint encoding so we can use integer atomicMax ----
__device__ __forceinline__ unsigned f2o(float f) {
    unsigned b = __float_as_uint(f);
    return (b & 0x80000000u) ? ~b : (b | 0x80000000u);
}
__device__ __forceinline__ float o2f(unsigned o) {
    unsigned b = (o & 0x80000000u) ? (o & 0x7FFFFFFFu) : ~o;
    return __uint_as_float(b);
}
// f2o(-inf) == 0x007FFFFFu  (used as init value)

// ---------------------------------------------------------------
// Kernel 0: init scratch accumulators
// ---------------------------------------------------------------
__global__ __launch_bounds__(256) void init_kernel(float* __restrict__ accum,
                                                   float* __restrict__ denom,
                                                   unsigned* __restrict__ amax) {
    int i = blockIdx.x * 256 + threadIdx.x;
    if (i < N_NODES * HC) accum[i] = 0.0f;
    if (i < N_NODES * HEADS) { denom[i] = 0.0f; amax[i] = 0x007FFFFFu; }
}

// ---------------------------------------------------------------
// Kernel 1: node GEMMs  x_l = x@W_l+b_l, x_r = x@W_r+b_r, xl = tanh(x@W1+b1)
// one block = 16 node rows, 8 waves, each wave does 6 of 48 col tiles
// ---------------------------------------------------------------
__global__ __launch_bounds__(256) void node_gemm_kernel(
    const float* __restrict__ x,
    const float* __restrict__ W_l, const float* __restrict__ b_l,
    const float* __restrict__ W_r, const float* __restrict__ b_r,
    const float* __restrict__ W1,  const float* __restrict__ b1,
    float* __restrict__ x_l, float* __restrict__ x_r, float* __restrict__ xl)
{
    const int wave = threadIdx.x >> 5;
    const int lane = threadIdx.x & 31;
    const int lr = lane & 15, hi = lane >> 4;
    const int r0 = blockIdx.x * 16;

    for (int t = wave; t < 48; t += 8) {
        const int o  = t >> 4;          // 0: W_l, 1: W_r, 2: W1
        const int c0 = (t & 15) << 4;   // col tile base
        const float* W = (o == 0) ? W_l : (o == 1) ? W_r : W1;
        const float* bias_p = (o == 0) ? b_l : (o == 1) ? b_r : b1;
        float* dst = (o == 0) ? x_l : (o == 1) ? x_r : xl;

        v8f acc = {};
        for (int k = 0; k < HID; k += 4) {
            const int ka = k + 2 * hi;
            v2f a, b;
            a.x = x[(r0 + lr) * HID + ka];
            a.y = x[(r0 + lr) * HID + ka + 1];
            b.x = W[ka * HC + c0 + lr];
            b.y = W[(ka + 1) * HC + c0 + lr];
            acc = __builtin_amdgcn_wmma_f32_16x16x4_f32(false, a, false, b,
                                                        (short)0, acc, false, false);
        }
        const float bias = bias_p[c0 + lr];
        #pragma unroll
        for (int i = 0; i < 8; ++i) {
            float v = acc[i] + bias;
            if (o == 2) v = tanhf(v);
            dst[(r0 + i + 8 * hi) * HC + c0 + lr] = v;
        }
    }
}

// ---------------------------------------------------------------
// Kernel 2: per-edge attention logits (fused e = edge_attr@W_e)
// one block = 16 edges; WMMA over K=32, fused gather + leakyrelu into LDS,
// then att-dot -> logit, atomicMax segment max
// ---------------------------------------------------------------
__global__ __launch_bounds__(256) void edge_attn_kernel(
    const long long* __restrict__ ei,
    const float* __restrict__ edge_attr,
    const float* __restrict__ W_e,
    const float* __restrict__ att,
    const float* __restrict__ x_l, const float* __restrict__ x_r,
    float* __restrict__ alpha, unsigned* __restrict__ amax)
{
    __shared__ float m[16][HC];     // leaky-relu'd messages, 16 KB
    __shared__ float red[256];
    __shared__ int s_src[16], s_dst[16];

    const int tid = threadIdx.x;
    const int wave = tid >> 5, lane = tid & 31;
    const int lr = lane & 15, hi = lane >> 4;
    const long long e0 = (long long)blockIdx.x * 16;

    if (tid < 16) {
        s_src[tid] = (int)ei[e0 + tid];
        s_dst[tid] = (int)ei[N_EDGES + e0 + tid];
    }
    __syncthreads();

    #pragma unroll
    for (int tt = 0; tt < 2; ++tt) {
        const int c0 = (wave + 8 * tt) << 4;
        v8f acc = {};
        for (int k = 0; k < EDIM; k += 4) {
            const int ka = k + 2 * hi;
            v2f a, b;
            a.x = edge_attr[(e0 + lr) * EDIM + ka];
            a.y = edge_attr[(e0 + lr) * EDIM + ka + 1];
            b.x = W_e[ka * HC + c0 + lr];
            b.y = W_e[(ka + 1) * HC + c0 + lr];
            acc = __builtin_amdgcn_wmma_f32_16x16x4_f32(false, a, false, b,
                                                        (short)0, acc, false, false);
        }
        #pragma unroll
        for (int i = 0; i < 8; ++i) {
            const int row = i + 8 * hi;       // edge within tile
            const int col = c0 + lr;
            float v = acc[i]
                    + x_l[(long long)s_src[row] * HC + col]
                    + x_r[(long long)s_dst[row] * HC + col];
            v = (v > 0.0f) ? v : NEG_SLOPE * v;
            m[row][col] = v;
        }
    }
    __syncthreads();

    // logit[edge][head] = sum_c m[edge][head*64+c]*att[head*64+c]
    // tid -> edge = tid>>4, head = (tid>>2)&3, sub = tid&3 (16-elem partial)
    const int edge = tid >> 4;
    const int head = (tid >> 2) & 3;
    const int sub  = tid & 3;
    const int cbase = head * HID + sub * 16;
    float partial = 0.0f;
    #pragma unroll
    for (int j = 0; j < 16; ++j)
        partial += m[edge][cbase + j] * att[cbase + j];
    red[tid] = partial;
    __syncthreads();

    if (sub == 0) {
        float logit = red[tid] + red[tid + 1] + red[tid + 2] + red[tid + 3];
        alpha[(e0 + edge) * HEADS + head] = logit;
        atomicMax(&amax[(long long)s_dst[edge] * HEADS + head], f2o(logit));
    }
}

// ---------------------------------------------------------------
// Kernel 3: alpha = exp(logit - amax[dst]); denom += alpha
// ---------------------------------------------------------------
__global__ __launch_bounds__(256) void softmax_denom_kernel(
    const long long* __restrict__ ei,
    float* __restrict__ alpha,
    const unsigned* __restrict__ amax,
    float* __restrict__ denom)
{
    int idx = blockIdx.x * 256 + threadIdx.x;
    if (idx >= N_EDGES * HEADS) return;
    int e = idx >> 2, h = idx & 3;
    int d = (int)ei[N_EDGES + e];
    float a = expf(alpha[idx] - o2f(amax[d * HEADS + h]));
    alpha[idx] = a;
    atomicAdd(&denom[d * HEADS + h], a);
}

// ---------------------------------------------------------------
// Kernel 4: accum[dst] += x_l[src] * alpha_unnorm  (one wave per edge)
// normalization by denom deferred to finalize
// ---------------------------------------------------------------
__global__ __launch_bounds__(256) void aggregate_kernel(
    const long long* __restrict__ ei,
    const float* __restrict__ x_l,
    const float* __restrict__ alpha,
    float* __restrict__ accum)
{
    const int wave = threadIdx.x >> 5, lane = threadIdx.x & 31;
    const long long e = (long long)blockIdx.x * 8 + wave;
    const int src = (int)ei[e];
    const int dst = (int)ei[N_EDGES + e];
    #pragma unroll
    for (int j = 0; j < 8; ++j) {
        const int c = lane + 32 * j;
        const float a = alpha[e * HEADS + (c >> 6)];
        atomicAdd(&accum[(long long)dst * HC + c],
                  x_l[(long long)src * HC + c] * a);
    }
}

// ---------------------------------------------------------------
// Kernel 5: finalize — gat out + residual + LayerNorm + z@W2 + tanh
// one block = 16 nodes; LN via wave32 shuffles; final GEMM via WMMA from LDS
// ---------------------------------------------------------------
__global__ __launch_bounds__(256) void finalize_kernel(
    const float* __restrict__ accum, const float* __restrict__ denom,
    const float* __restrict__ bias_gat,
    const float* __restrict__ xl,
    const float* __restrict__ gamma, const float* __restrict__ beta,
    const float* __restrict__ W2, const float* __restrict__ b2,
    float* __restrict__ out)
{
    __shared__ float z[16][HC];   // normalized rows, 16 KB
    const int wave = threadIdx.x >> 5, lane = threadIdx.x & 31;
    const int n0 = blockIdx.x * 16;

    #pragma unroll
    for (int t = 0; t < 2; ++t) {
        const int nl = wave + 8 * t;
        const long long n = n0 + nl;
        float vals[8];
        float sum = 0.0f, sumsq = 0.0f;
        #pragma unroll
        for (int j = 0; j < 8; ++j) {
            const int c = lane + 32 * j;
            float hg = accum[n * HC + c] / (denom[n * HEADS + (c >> 6)] + 1e-16f)
                     + bias_gat[c];
            float v = xl[n * HC + c] + hg;
            vals[j] = v; sum += v; sumsq += v * v;
        }
        #pragma unroll
        for (int o = 16; o > 0; o >>= 1) {
            sum   += __shfl_xor(sum, o, 32);
            sumsq += __shfl_xor(sumsq, o, 32);
        }
        const float mean = sum * (1.0f / HC);
        const float var  = sumsq * (1.0f / HC) - mean * mean;
        const float rstd = rsqrtf(var + LN_EPS);
        #pragma unroll
        for (int j = 0; j < 8; ++j) {
            const int c = lane + 32 * j;
            z[nl][c] = (vals[j] - mean) * rstd * gamma[c] + beta[c];
        }
    }
    __syncthreads();

    if (wave < 4) {   // 4 col tiles of 16 (HID=64)
        const int lr = lane & 15, hi = lane >> 4;
        const int c0 = wave << 4;
        v8f acc = {};
        for (int k = 0; k < HC; k += 4) {
            const int ka = k + 2 * hi;
            v2f a, b;
            a.x = z[lr][ka];
            a.y = z[lr][ka + 1];
            b.x = W2[ka * HID + c0 + lr];
            b.y = W2[(ka + 1) * HID + c0 + lr];
            acc = __builtin_amdgcn_wmma_f32_16x16x4_f32(false, a, false, b,
                                                        (short)0, acc, false, false);
        }
        #pragma unroll
        for (int i = 0; i < 8; ++i)
            out[(long long)(n0 + i + 8 * hi) * HID + c0 + lr] =
                tanhf(acc[i] + b2[c0 + lr]);
    }
}

// ---------------------------------------------------------------
extern "C" void kernel_launch(void* const* d_in, const int* in_sizes, int n_in,
                              void* d_out, int out_size, void* d_ws, size_t ws_size,
                              hipStream_t stream) {
    const float*      x        = (const float*)d_in[0];
    const long long*  ei       = (const long long*)d_in[1];
    const float*      edge_attr= (const float*)d_in[2];
    const float*      W_l      = (const float*)d_in[3];
    const float*      b_l      = (const float*)d_in[4];
    const float*      W_r      = (const float*)d_in[5];
    const float*      b_r      = (const float*)d_in[6];
    const float*      W_e      = (const float*)d_in[7];
    const float*      att      = (const float*)d_in[8];
    const float*      bias_gat = (const float*)d_in[9];
    const float*      W1       = (const float*)d_in[10];
    const float*      b1       = (const float*)d_in[11];
    const float*      gamma    = (const float*)d_in[12];
    const float*      beta     = (const float*)d_in[13];
    const float*      W2       = (const float*)d_in[14];
    const float*      b2       = (const float*)d_in[15];
    float* out = (float*)d_out;

    // workspace layout
    float* ws    = (float*)d_ws;
    float* x_l   = ws;                                   // N*HC
    float* x_r   = x_l   + (size_t)N_NODES * HC;         // N*HC
    float* xl    = x_r   + (size_t)N_NODES * HC;         // N*HC
    float* accum = xl    + (size_t)N_NODES * HC;         // N*HC
    unsigned* amax = (unsigned*)(accum + (size_t)N_NODES * HC); // N*HEADS
    float* denom = (float*)(amax + (size_t)N_NODES * HEADS);    // N*HEADS
    float* alpha = denom + (size_t)N_NODES * HEADS;             // E*HEADS

    init_kernel<<<(N_NODES * HC + 255) / 256, 256, 0, stream>>>(accum, denom, amax);
    node_gemm_kernel<<<N_NODES / 16, 256, 0, stream>>>(x, W_l, b_l, W_r, b_r, W1, b1,
                                                       x_l, x_r, xl);
    edge_attn_kernel<<<N_EDGES / 16, 256, 0, stream>>>(ei, edge_attr, W_e, att,
                                                       x_l, x_r, alpha, amax);
    softmax_denom_kernel<<<(N_EDGES * HEADS + 255) / 256, 256, 0, stream>>>(ei, alpha,
                                                                            amax, denom);
    aggregate_kernel<<<N_EDGES / 8, 256, 0, stream>>>(ei, x_l, alpha, accum);
    finalize_kernel<<<N_NODES / 16, 256, 0, stream>>>(accum, denom, bias_gat, xl,
                                                      gamma, beta, W2, b2, out);
}